// PQC_26757646254573
// MI455X (gfx1250) — compile-verified
//
#include <hip/hip_runtime.h>
#include <hip/hip_bf16.h>

// ---------------------------------------------------------------------------
// PQC simulator for gfx1250 (MI455X).
//  Stage 1: build per-qubit 2x2 gates M = Rz*Rx                (96 threads)
//  Stage 2: build dense H = h^{(x)8} (re plane, zero im)        (elementwise)
//  Stage 3: build d_l = C0 r C1 r ... C6 r row-wise, one wave32 per row,
//           register-resident with __shfl_xor butterflies (no LDS needed)
//  Stage 4: product tree U = ((d0 d1)(d2 d3)) ((d4 d5) H) via complex
//           256x256x256 GEMMs: V_WMMA_F32_16X16X4_F32 fed from LDS, with
//           TENSOR_LOAD_TO_LDS (TDM) double-buffered K-chunk staging
//  Stage 5: D = zdiag * |U|^2 ; out[b] = exp(f1 + i f2)
// ---------------------------------------------------------------------------

#define INVC 0.87055056329612413f   /* 2^-0.2 */

typedef __attribute__((ext_vector_type(2))) float        v2f;
typedef __attribute__((ext_vector_type(8))) float        v8f;
typedef __attribute__((ext_vector_type(4))) unsigned int v4u;
typedef __attribute__((ext_vector_type(4))) int          v4i;
typedef __attribute__((ext_vector_type(8))) int          v8i;

#if defined(__has_builtin)
#if __has_builtin(__builtin_amdgcn_tensor_load_to_lds) && \
    __has_builtin(__builtin_amdgcn_s_wait_tensorcnt)
#define HAVE_TDM 1
#endif
#endif

struct cf { float re, im; };
__device__ __forceinline__ cf cmul(cf a, cf b) {
    return { a.re * b.re - a.im * b.im, a.re * b.im + a.im * b.re };
}
__device__ __forceinline__ cf cadd(cf a, cf b) { return { a.re + b.re, a.im + b.im }; }
__device__ __forceinline__ cf cshfl_xor(cf v, int m) {
    return { __shfl_xor(v.re, m, 32), __shfl_xor(v.im, m, 32) };
}

// ---- Stage 1: 2x2 gates: M = Rz(qz) * Rx(qx), stored (re,im)x4 ------------
__global__ void k_gates(const float* qx1, const float* qz1,
                        const float* qx2, const float* qz2, float* g) {
    int t = blockIdx.x * blockDim.x + threadIdx.x;
    if (t >= 96) return;                       // 2 sets * 6 layers * 8 qubits
    int s = t / 48, r = t % 48, l = r / 8, q = r % 8;
    const float* qx = s ? qx2 : qx1;
    const float* qz = s ? qz2 : qz1;
    float ax = qx[l * 8 + q] * 0.5f, az = qz[l * 8 + q] * 0.5f;
    float cx = cosf(ax), sx = sinf(ax), cz = cosf(az), sz = sinf(az);
    float* o = g + t * 8;
    o[0] =  cz * cx;  o[1] =  sz * cx;    // M00 = (cz+i sz) cx
    o[2] = -sz * sx;  o[3] =  cz * sx;    // M01 = (cz+i sz)(i sx)
    o[4] =  sz * sx;  o[5] =  cz * sx;    // M10 = (cz-i sz)(i sx)
    o[6] =  cz * cx;  o[7] = -sz * cx;    // M11 = (cz-i sz) cx
}

// ---- Stage 2: Hadamard^{(x)8}: H[i][j] = (-1)^popc(i&j)/16 ----------------
__global__ void k_H(float* Hre) {
    int t = blockIdx.x * 256 + threadIdx.x;   // 65536 threads
    int i = t >> 8, j = t & 255;
    Hre[t]         = (__popc(i & j) & 1) ? -0.0625f : 0.0625f;
    Hre[65536 + t] = 0.0f;                    // zero imaginary plane
}

// ---- Stage 3: d_l rows, one wave per row, 8 complex entries per lane ------
__global__ void k_build_d(const float* g, float* d_base) {
    int wid  = (blockIdx.x * blockDim.x + threadIdx.x) >> 5;
    int lane = threadIdx.x & 31;
    int s = wid / (6 * 256);
    int r = wid % (6 * 256);
    int l = r >> 8;
    int row = r & 255;

    cf a[8];
    for (int t = 0; t < 8; t++) {             // init: row of C0 (ctrl=128, tgt=64)
        int j = t * 32 + lane;
        cf v = { 0.f, 0.f };
        if (j == row) v = (row & 128) ? cf{ INVC, 2.f * INVC } : cf{ INVC, 0.f };
        if ((row & 128) && j == (row ^ 64)) v = { 0.f, -2.f * INVC };
        a[t] = v;
    }

    for (int n = 0; n < 7; n++) {
        for (int q = 0; q < 8; q++) {         // apply r_l (8 qubit butterflies)
            int mask = 1 << (7 - q);
            const float* mg = g + ((s * 6 + l) * 8 + q) * 8;
            cf M00 = { mg[0], mg[1] }, M01 = { mg[2], mg[3] };
            cf M10 = { mg[4], mg[5] }, M11 = { mg[6], mg[7] };
            cf na[8];
            for (int t = 0; t < 8; t++) {
                int j = t * 32 + lane;
                cf p = (mask >= 32) ? a[t ^ (mask >> 5)] : cshfl_xor(a[t], mask);
                if (j & mask) na[t] = cadd(cmul(p, M01), cmul(a[t], M11));
                else          na[t] = cadd(cmul(a[t], M00), cmul(p, M10));
            }
            for (int t = 0; t < 8; t++) a[t] = na[t];
        }
        if (n < 6) {                          // apply C_{n+1}
            int nn = n + 1;
            int cmask = 1 << (7 - nn), tmask = 1 << (6 - nn);
            cf na[8];
            for (int t = 0; t < 8; t++) {
                int j = t * 32 + lane;
                cf p = (tmask >= 32) ? a[t ^ (tmask >> 5)] : cshfl_xor(a[t], tmask);
                if (j & cmask) {
                    na[t].re = (a[t].re - 2.f * a[t].im + 2.f * p.im) * INVC;
                    na[t].im = (a[t].im + 2.f * a[t].re - 2.f * p.re) * INVC;
                } else {
                    na[t].re = a[t].re * INVC;
                    na[t].im = a[t].im * INVC;
                }
            }
            for (int t = 0; t < 8; t++) a[t] = na[t];
        }
    }

    float* dre = d_base + (size_t)(s * 6 + l) * 2 * 65536;
    float* dim_ = dre + 65536;
    for (int t = 0; t < 8; t++) {
        int j = t * 32 + lane;
        dre[row * 256 + j]  = a[t].re;
        dim_[row * 256 + j] = a[t].im;
    }
}

// ---- Stage 4: batched complex GEMM, WMMA f32 16x16x4 + TDM LDS staging ----
// Matrix layout: [re plane 65536][im plane 65536], row-major 256x256.
// Block: 32-row x 64-col output tile, 8 waves (2x4 of 16x16 tiles).
// K chunked by 32, double buffered. Per chunk one TDM tile per operand moves
// BOTH planes (z-dim = plane, stride 65536 elements).
struct CGemmBatch {
    const float* A[6];   // re plane; im = A + 65536
    const float* B[6];
    float*       C[6];
};

#define AS_RSTRIDE 36                 // 32 K-floats + 4 pad (TDM pad_enable)
#define AS_PSTRIDE (32 * AS_RSTRIDE)  // 1152 floats per plane
#define AS_BUF     (2 * AS_PSTRIDE)   // 2304 floats per buffer
#define BS_PSTRIDE (32 * 64)          // 2048 floats per plane
#define BS_BUF     (2 * BS_PSTRIDE)   // 4096 floats per buffer

__device__ __forceinline__ unsigned lds_off(const void* p) {
    return (unsigned)(unsigned long long)(uintptr_t)p;   // flat addr[31:0] = LDS offset
}

#ifdef HAVE_TDM
// 3-D TDM tile: x = tile0 contiguous floats, y = tile1 rows (stride0=256),
// z = 2 planes (stride1=65536). w0flags carries data_size + pad config.
__device__ __forceinline__ void tdm_load(unsigned ldsaddr, const float* gptr,
                                         unsigned w0flags, unsigned tile0,
                                         unsigned tile1) {
    unsigned long long ga = (unsigned long long)(uintptr_t)gptr;
    const unsigned td0 = 1u << 20, td1 = 1u << 20;   // oversized dims: no OOB clip
    v4u g0;
    g0[0] = 1u;                                       // count=1, user descriptor
    g0[1] = ldsaddr;                                  // lds_addr (bytes)
    g0[2] = (unsigned)ga;                             // global_addr[31:0]
    g0[3] = (unsigned)((ga >> 32) & 0x1FFFFFFull) | (2u << 30);  // addr hi | type=2
    v8i g1;
    g1[0] = (int)w0flags;                             // data_size=4B (+pad cfg)
    g1[1] = (int)((td0 & 0xFFFFu) << 16);             // tensor_dim0 lo16
    g1[2] = (int)((td0 >> 16) | ((td1 & 0xFFFFu) << 16));
    g1[3] = (int)((td1 >> 16) | (tile0 << 16));       // tile_dim0
    g1[4] = (int)(tile1 | (2u << 16));                // tile_dim1 | tile_dim2=2
    g1[5] = 256;                                      // tensor_dim0_stride lo32
    g1[6] = 0;                                        // stride0 hi | stride1 lo16 (65536 -> 0)
    g1[7] = 1;                                        // stride1 bits[47:16] = 1 (65536)
    v4i g2; g2[0] = 4; g2[1] = 0; g2[2] = 0; g2[3] = 0;  // tensor_dim2=4, rest off
    v4i g3; g3[0] = 0; g3[1] = 0; g3[2] = 0; g3[3] = 0;
#if __clang_major__ >= 23
    v8i gx; gx[0]=0; gx[1]=0; gx[2]=0; gx[3]=0; gx[4]=0; gx[5]=0; gx[6]=0; gx[7]=0;
    __builtin_amdgcn_tensor_load_to_lds(g0, g1, g2, g3, gx, 0);
#else
    __builtin_amdgcn_tensor_load_to_lds(g0, g1, g2, g3, 0);
#endif
}
#endif

__global__ void k_cgemm(CGemmBatch gb) {
    __shared__ float sA[2][AS_BUF];   // [buf][plane][32 rows][36]   (~18.4 KB)
    __shared__ float sB[2][BS_BUF];   // [buf][plane][32 rows][64]   (~32.8 KB)

    int gi = blockIdx.y;
    const float* A = gb.A[gi];
    const float* B = gb.B[gi];
    float*       C = gb.C[gi];

    int rb   = blockIdx.x >> 2;       // 0..7 : 32-row block
    int cb   = blockIdx.x & 3;        // 0..3 : 64-col block
    int w    = threadIdx.x >> 5;      // wave 0..7
    int lane = threadIdx.x & 31;
    int wr = w >> 2, wc = w & 3;      // wave tile in 2x4 arrangement
    int half = lane >> 4, m16 = lane & 15;
    int lr = wr * 16 + m16;           // A row within block (0..31)
    int lc = wc * 16 + m16;           // B col within block (0..63)

    auto load_chunk = [&](int buf, int k0) {
#ifdef HAVE_TDM
        if (threadIdx.x < 32) {       // one wave issues the DMA; EXEC ignored
            // A panel: 32 K-floats x 32 rows x 2 planes, pad 4 DW per 32-DW row
            //   w0: data_size=2(4B) | pad_en | pad_interval=4(32DW) | pad_amt=3(4DW)
            tdm_load(lds_off(&sA[buf][0]), A + rb * 32 * 256 + k0,
                     0x20000u | (1u << 20) | (4u << 22) | (3u << 25), 32u, 32u);
            // B panel: 64 cols x 32 K-rows x 2 planes, no padding
            tdm_load(lds_off(&sB[buf][0]), B + k0 * 256 + cb * 64,
                     0x20000u, 64u, 32u);
        }
#else
        for (int e = threadIdx.x; e < 2 * 32 * 32; e += 256) {
            int p = e >> 10, rr = (e >> 5) & 31, kk = e & 31;
            sA[buf][p * AS_PSTRIDE + rr * AS_RSTRIDE + kk] =
                A[p * 65536 + (rb * 32 + rr) * 256 + k0 + kk];
        }
        for (int e = threadIdx.x; e < 2 * 32 * 64; e += 256) {
            int p = e >> 11, rr = (e >> 6) & 31, cc = e & 63;
            sB[buf][p * BS_PSTRIDE + rr * 64 + cc] =
                B[p * 65536 + (k0 + rr) * 256 + cb * 64 + cc];
        }
#endif
    };

    load_chunk(0, 0);
    v8f accr = {}; v8f acci = {};

    for (int kc = 0; kc < 8; kc++) {
        int buf = kc & 1;
        if (kc < 7) load_chunk(buf ^ 1, (kc + 1) * 32);
#ifdef HAVE_TDM
        if (kc < 7) __builtin_amdgcn_s_wait_tensorcnt(2);  // chunk kc landed
        else        __builtin_amdgcn_s_wait_tensorcnt(0);
#endif
        __syncthreads();

        const float* a0 = &sA[buf][lr * AS_RSTRIDE];               // re
        const float* a1 = &sA[buf][AS_PSTRIDE + lr * AS_RSTRIDE];  // im
        const float* b0 = &sB[buf][0];
        const float* b1 = &sB[buf][BS_PSTRIDE];

        for (int k = 0; k < 32; k += 4) {
            int ka = k + 2 * half;
            v2f ar = *(const v2f*)(a0 + ka);      // A frag: K=ka, ka+1
            v2f ai = *(const v2f*)(a1 + ka);
            v2f br, bi;
            br[0] = b0[(ka + 0) * 64 + lc]; br[1] = b0[(ka + 1) * 64 + lc];
            bi[0] = b1[(ka + 0) * 64 + lc]; bi[1] = b1[(ka + 1) * 64 + lc];
            v2f nai; nai[0] = -ai[0]; nai[1] = -ai[1];
            // Cr += Ar*Br - Ai*Bi ; Ci += Ar*Bi + Ai*Br
            accr = __builtin_amdgcn_wmma_f32_16x16x4_f32(false, ar,  false, br, (short)0, accr, false, false);
            accr = __builtin_amdgcn_wmma_f32_16x16x4_f32(false, nai, false, bi, (short)0, accr, false, false);
            acci = __builtin_amdgcn_wmma_f32_16x16x4_f32(false, ar,  false, bi, (short)0, acci, false, false);
            acci = __builtin_amdgcn_wmma_f32_16x16x4_f32(false, ai,  false, br, (short)0, acci, false, false);
        }
        __syncthreads();                          // buffer free for next DMA
    }

    int row0 = rb * 32 + wr * 16, col0 = cb * 64 + wc * 16;
    for (int v = 0; v < 8; v++) {                 // C/D: VGPR v -> M = v + 8*half
        int m = row0 + v + 8 * half;
        int n = col0 + m16;
        C[m * 256 + n]         = accr[v];
        C[65536 + m * 256 + n] = acci[v];
    }
}

// ---- Stage 5a: D[q][j] = sum_i (+-1)^{bit_q(i)} * |U[i][j]|^2 -------------
__global__ void k_reduce(const float* U0, const float* U1, float* Dm) {
    int s = blockIdx.x;
    int j = threadIdx.x;                      // 256 columns
    const float* Ur = s ? U1 : U0;
    const float* Ui = Ur + 65536;
    float acc[8] = {0.f,0.f,0.f,0.f,0.f,0.f,0.f,0.f};
    for (int i = 0; i < 256; i++) {
        float ur = Ur[i * 256 + j], ui = Ui[i * 256 + j];
        float u2 = ur * ur + ui * ui;
        for (int q = 0; q < 8; q++)
            acc[q] += ((i >> (7 - q)) & 1) ? -u2 : u2;
    }
    for (int q = 0; q < 8; q++) Dm[s * 2048 + q * 256 + j] = acc[q];
}

// ---- Stage 5b: out[b] = exp(f1 + i f2), interleaved (re,im) float ---------
__global__ void k_output(const float* x, int batch, const float* Dm,
                         float* out, int out_size) {
    int b = blockIdx.x * blockDim.x + threadIdx.x;
    if (b >= batch) return;
    int idx = 0, idxl = 0;
    for (int q = 0; q < 8; q++) {
        idx  |= (x[b * 8 + q]           > 0.f) << (7 - q);
        idxl |= (x[(batch - 1) * 8 + q] > 0.f) << (7 - q);
    }
    float f1 = (idxl < 8) ? Dm[0 * 2048 + idxl * 256 + idx] : 0.f;
    float f2 = (idxl < 8) ? Dm[1 * 2048 + idxl * 256 + idx] : 0.f;
    float ef = expf(f1);
    if (2 * b     < out_size) out[2 * b]     = ef * cosf(f2);
    if (2 * b + 1 < out_size) out[2 * b + 1] = ef * sinf(f2);
}

// ---------------------------------------------------------------------------
extern "C" void kernel_launch(void* const* d_in, const int* in_sizes, int n_in,
                              void* d_out, int out_size, void* d_ws, size_t ws_size,
                              hipStream_t stream) {
    const float* x   = (const float*)d_in[0];
    const float* qx1 = (const float*)d_in[1];
    const float* qz1 = (const float*)d_in[2];
    const float* qx2 = (const float*)d_in[3];
    const float* qz2 = (const float*)d_in[4];
    // d_in[5], d_in[6] (c1, c2) are unused by the reference math.
    int batch = in_sizes[0] / 8;

    // Workspace layout (floats). Total ~= 8192 + 25*131072 floats ~= 13.1 MB.
    float* ws = (float*)d_ws;
    float* g      = ws;            // 768   : 2x6x8 gates, (re,im)x4
    float* Dm     = ws + 1024;     // 4096  : D matrices, 2 sets x 8 x 256
    float* planes = ws + 8192;     // matrix slots: [re 64K][im 64K] each
    auto sr = [&](int i) { return planes + (size_t)i * 131072; };
    // slot 0 = H ; 1+s*6+l = d[s][l] ; 13+s*3+p = P ; 19+s*2+q = Q ; 23+s = U

    k_gates  <<<1,   96, 0, stream>>>(qx1, qz1, qx2, qz2, g);
    k_H      <<<256, 256, 0, stream>>>(sr(0));
    k_build_d<<<384, 256, 0, stream>>>(g, sr(1));   // writes slots 1..12

    CGemmBatch b1{};                                 // P = d*d (6 gemms)
    for (int s = 0; s < 2; s++)
        for (int p = 0; p < 3; p++) {
            int gi = s * 3 + p;
            int a = 1 + s * 6 + 2 * p;
            b1.A[gi] = sr(a);
            b1.B[gi] = sr(a + 1);
            b1.C[gi] = sr(13 + s * 3 + p);
        }
    k_cgemm<<<dim3(32, 6), 256, 0, stream>>>(b1);

    CGemmBatch b2{};                                 // Q0 = P0*P1 ; Q1 = P2*H
    for (int s = 0; s < 2; s++) {
        b2.A[s * 2 + 0] = sr(13 + s * 3 + 0);
        b2.B[s * 2 + 0] = sr(13 + s * 3 + 1);
        b2.C[s * 2 + 0] = sr(19 + s * 2 + 0);
        b2.A[s * 2 + 1] = sr(13 + s * 3 + 2);
        b2.B[s * 2 + 1] = sr(0);
        b2.C[s * 2 + 1] = sr(19 + s * 2 + 1);
    }
    k_cgemm<<<dim3(32, 4), 256, 0, stream>>>(b2);

    CGemmBatch b3{};                                 // U = Q0*Q1
    for (int s = 0; s < 2; s++) {
        b3.A[s] = sr(19 + s * 2 + 0);
        b3.B[s] = sr(19 + s * 2 + 1);
        b3.C[s] = sr(23 + s);
    }
    k_cgemm<<<dim3(32, 2), 256, 0, stream>>>(b3);

    k_reduce <<<2, 256, 0, stream>>>(sr(23), sr(24), Dm);
    k_output <<<(batch + 255) / 256, 256, 0, stream>>>(x, batch, Dm,
                                                       (float*)d_out, out_size);
    (void)n_in; (void)ws_size;
}